// BiLSTM_CRF_78804059947030
// MI455X (gfx1250) — compile-verified
//
#include <hip/hip_runtime.h>

typedef __attribute__((ext_vector_type(2))) float v2f;
typedef __attribute__((ext_vector_type(8))) float v8f;

#define K_TAGS   32
#define START_IX 30
#define END_IX   31

// One wave32 per block; each wave advances 16 sequences through the CRF scan.
// Per time-step: 16x v_wmma_f32_16x16x4_f32 compute s = a * exp(trans),
// then log / rowmax / exp renormalization in f32 VALU.
__global__ __launch_bounds__(32) void crf_forward_wmma(
    const float* __restrict__ frames,   // [B, T, 32]
    const float* __restrict__ trans,    // [32, 32] prev-major
    float* __restrict__ out,            // [B]
    int B, int T)
{
  __shared__ float lds_a[16 * K_TAGS];  // a[m][k], m = batch row in group
  __shared__ float lds_L[16];

  const int lane = threadIdx.x;   // 0..31
  const int lo   = lane & 15;
  const int hi   = lane >> 4;     // 0 or 1 (lane half)
  const int b0   = blockIdx.x * 16;

  // ---- constant B operand: E[k][n] = exp(trans[k][n]) in f32 4x16 WMMA layout.
  // chunk c covers k = 4c..4c+3 ; half h covers n = 16h..16h+15.
  // lane holds k = 4c + 2*hi (+0/+1 in .x/.y), column n = 16h + lo.
  v2f Bc[8][2];
#pragma unroll
  for (int c = 0; c < 8; ++c) {
    const int k0 = 4 * c + 2 * hi;
#pragma unroll
    for (int h = 0; h < 2; ++h) {
      const int n = 16 * h + lo;
      v2f e;
      e.x = __expf(trans[(k0 + 0) * K_TAGS + n]);   // exp(-10000) -> exact 0
      e.y = __expf(trans[(k0 + 1) * K_TAGS + n]);
      Bc[c][h] = e;
    }
  }

  // ---- per-row frame base pointers (D layout: VGPR r <-> batch row r + 8*hi)
  const float* pr[8];
#pragma unroll
  for (int r = 0; r < 8; ++r) {
    int row = b0 + r + 8 * hi;
    if (row >= B) row = B - 1;                       // branchless clamp, EXEC stays full
    pr[r] = frames + (size_t)row * (size_t)T * K_TAGS + lo;
  }

  // ---- init: alpha0 = 0 at START, -inf elsewhere  =>  a = one-hot(START), L = 0
#pragma unroll
  for (int i = 0; i < 16; ++i) {
    const int idx = lane + 32 * i;                   // idx = m*32 + k
    lds_a[idx] = ((idx & 31) == START_IX) ? 1.0f : 0.0f;
  }

  // A operand (16x4 f32 layout): lane holds row m=lo, k = 4c + 2*hi (+0/+1)
  v2f A[8];
#pragma unroll
  for (int c = 0; c < 8; ++c)
    A[c] = *(const v2f*)&lds_a[lo * K_TAGS + 4 * c + 2 * hi];

  float Lacc[8];
#pragma unroll
  for (int r = 0; r < 8; ++r) Lacc[r] = 0.0f;

  // frames for t=0, D layout: f0 -> columns 0..15, f1 -> columns 16..31
  float f0[8], f1[8];
#pragma unroll
  for (int r = 0; r < 8; ++r) { f0[r] = pr[r][0]; f1[r] = pr[r][16]; }

  for (int t = 0; t < T; ++t) {
    // -------- prefetch frames for t+1 (clamped, branchless)
    const int tp = (t + 1 < T) ? (t + 1) : t;
    float g0[8], g1[8];
#pragma unroll
    for (int r = 0; r < 8; ++r) {
      g0[r] = pr[r][tp * K_TAGS];
      g1[r] = pr[r][tp * K_TAGS + 16];
    }

    // -------- s[m][n] = sum_k a[m][k] * E[k][n]  : 16x v_wmma_f32_16x16x4_f32
    v8f acc0 = {}; v8f acc1 = {};
#pragma unroll
    for (int c = 0; c < 8; ++c) {
      acc0 = __builtin_amdgcn_wmma_f32_16x16x4_f32(false, A[c], false, Bc[c][0],
                                                   (short)0, acc0, false, false);
      acc1 = __builtin_amdgcn_wmma_f32_16x16x4_f32(false, A[c], false, Bc[c][1],
                                                   (short)0, acc1, false, false);
    }

    // -------- v = frame + log(s); per-row max over all 32 columns
    float v0[8], v1[8], w[8];
#pragma unroll
    for (int r = 0; r < 8; ++r) {
      v0[r] = f0[r] + __logf(acc0[r]);   // log(0) = -inf for dead columns (e.g. START)
      v1[r] = f1[r] + __logf(acc1[r]);
      w[r]  = fmaxf(v0[r], v1[r]);
    }
#pragma unroll
    for (int m = 1; m <= 8; m <<= 1) {   // reduce across the 16 lanes of each half
#pragma unroll
      for (int r = 0; r < 8; ++r)
        w[r] = fmaxf(w[r], __shfl_xor(w[r], m, 32));
    }
#pragma unroll
    for (int r = 0; r < 8; ++r) Lacc[r] += w[r];

    // -------- a = exp(v - max): store in LDS (D layout -> a[m][k]), reload in A layout
#pragma unroll
    for (int r = 0; r < 8; ++r) {
      const int m = r + 8 * hi;
      lds_a[m * K_TAGS + lo]      = __expf(v0[r] - w[r]);
      lds_a[m * K_TAGS + 16 + lo] = __expf(v1[r] - w[r]);
    }
#pragma unroll
    for (int c = 0; c < 8; ++c)
      A[c] = *(const v2f*)&lds_a[lo * K_TAGS + 4 * c + 2 * hi];

#pragma unroll
    for (int r = 0; r < 8; ++r) { f0[r] = g0[r]; f1[r] = g1[r]; }
  }

  // ---- publish L per batch row via LDS (DS ops are in-order within a wave)
#pragma unroll
  for (int r = 0; r < 8; ++r) {
    if (lane == 0)  lds_L[r]     = Lacc[r];
    if (lane == 16) lds_L[8 + r] = Lacc[r];
  }

  // ---- terminal transition into END: score = L + log(sum_k a[k] * exp(trans[k][END]))
  if (lane < 16) {
    const int b = b0 + lane;
    if (b < B) {
      float sum = 0.0f;
#pragma unroll
      for (int k = 0; k < K_TAGS; ++k)
        sum += lds_a[lane * K_TAGS + k] * __expf(trans[k * K_TAGS + END_IX]);
      out[b] = lds_L[lane] + __logf(sum);
    }
  }
}

extern "C" void kernel_launch(void* const* d_in, const int* in_sizes, int n_in,
                              void* d_out, int out_size, void* d_ws, size_t ws_size,
                              hipStream_t stream) {
  const float* frames = (const float*)d_in[0];
  const float* trans  = (const float*)d_in[1];
  float* out          = (float*)d_out;

  const int B = out_size;                                   // one score per sequence
  const long long total = (long long)in_sizes[0];
  const int T = (int)(total / ((long long)B * K_TAGS));

  const int blocks = (B + 15) / 16;                         // one wave per 16 sequences
  crf_forward_wmma<<<blocks, 32, 0, stream>>>(frames, trans, out, B, T);
}